// Mamba2Detection_32512902431054
// MI455X (gfx1250) — compile-verified
//
#include <hip/hip_runtime.h>
#include <hip/hip_bf16.h>

// CDNA5 / gfx1250 (MI455X), wave32.
// Live dataflow of the reference (z, dt, Bm, Cm are dead):
//   conv3x3(x, W_in[128:256])          -> A  (8,128,128,128)  [WMMA f32 16x16x4, async-LDS staged]
//   dwconv4x4(A, W_conv[0:128]) +b,silu-> xh (8,128,131,131)  [VALU]
//   per-32-float-group GEMM with W_out -> out (17,572,864 f32) [WMMA f32 16x16x4]

typedef __attribute__((ext_vector_type(2))) float v2f;
typedef __attribute__((ext_vector_type(8))) float v8f;

#define WMMA_F32X4(A_, B_, C_) \
  __builtin_amdgcn_wmma_f32_16x16x4_f32(false, (A_), false, (B_), (short)0, (C_), false, false)

// --- async global->LDS (ASYNCcnt) path, guarded: falls back to sync fill ----
#if defined(__has_builtin)
#if __has_builtin(__builtin_amdgcn_global_load_async_to_lds_b32) && \
    __has_builtin(__builtin_amdgcn_s_wait_asynccnt)
#define HAVE_ASYNC_LDS 1
#endif
#endif

#ifdef HAVE_ASYNC_LDS
typedef __attribute__((address_space(1))) int* gas_i32p;   // global AS pointer
typedef __attribute__((address_space(3))) int* las_i32p;   // LDS AS pointer
__device__ __forceinline__ void async_cp_b32(const float* g, float* l) {
  __builtin_amdgcn_global_load_async_to_lds_b32(
      (gas_i32p)g, (las_i32p)l, 0, 0);
}
#define ASYNC_WAIT() __builtin_amdgcn_s_wait_asynccnt(0)
#else
#define ASYNC_WAIT() ((void)0)
#endif

// ---------------------------------------------------------------------------
// Kernel 1: 3x3 conv, pad 1, input (8,64,128,128), weights rows 128..255 of
// W_in (388,64,3,3). GEMM view: M=out-channel (co), N=pixel, K=ci*9+ky*3+kx.
// Block = 128 threads (4 waves). Block tile: 64 co x 64 pixels (one row seg).
// Input channels processed in 8 chunks of 8; chunk c+1 streams into the
// other LDS buffer (async-to-LDS) while chunk c feeds 72 K-steps of WMMA.
// ---------------------------------------------------------------------------
#define CI_CHUNK    8
#define N_CHUNKS    8                      // 64 / CI_CHUNK
#define CHUNK_ELEMS (CI_CHUNK * 3 * 66)    // 1584 floats = 6336 B
#define CHUNK_K     (CI_CHUNK * 9)         // 72

__device__ __forceinline__ void fill_chunk(float* __restrict__ dst,
                                           const float* __restrict__ x,
                                           int ci0, int b, int y, int x0, int tid)
{
  for (int e = tid; e < CHUNK_ELEMS; e += 128) {
    int ci  = e / 198;                 // local channel within chunk
    int rem = e - ci * 198;
    int ry  = rem / 66;
    int cx  = rem - ry * 66;
    int iy  = y + ry - 1;
    int ix  = x0 + cx - 1;
    bool ok = (iy >= 0) & (iy < 128) & (ix >= 0) & (ix < 128);
    const float* g = x + (((size_t)(b * 64 + ci0 + ci)) << 14) + (iy << 7) + ix;
#ifdef HAVE_ASYNC_LDS
    if (ok)  async_cp_b32(g, &dst[e]);   // ASYNCcnt-tracked DMA into LDS
    else     dst[e] = 0.0f;              // zero halo via plain ds_store
#else
    float v = 0.0f;
    if (ok) v = *g;
    dst[e] = v;
#endif
  }
}

__launch_bounds__(128)
__global__ void conv1_wmma(const float* __restrict__ x,
                           const float* __restrict__ W_in,
                           float* __restrict__ A)
{
  __shared__ float lds[2 * CHUNK_ELEMS];   // 12672 B double buffer

  const int p0 = blockIdx.x * 64;      // first pixel of this block's row segment
  const int b  = p0 >> 14;             // /(128*128)
  const int y  = (p0 >> 7) & 127;
  const int x0 = p0 & 127;             // 0 or 64 (segments never cross rows)

  const int tid  = threadIdx.x;
  const int lane = tid & 31;
  const int wave = tid >> 5;
  const int half = lane >> 4;          // 0: K+{0,1}, 1: K+{2,3} (ISA A/B striping)
  const int l16  = lane & 15;

  const int coBase = blockIdx.y * 64 + wave * 16;    // local co tile base
  const int wrow   = (128 + coBase + l16) * 576;     // W_in row for A fragment
  __builtin_prefetch(&W_in[wrow], 0, 3);             // global_prefetch_b8

  v8f acc0 = {}, acc1 = {}, acc2 = {}, acc3 = {};

  fill_chunk(lds, x, 0, b, y, x0, tid);              // prime buffer 0

  for (int c = 0; c < N_CHUNKS; ++c) {
    ASYNC_WAIT();          // this wave's in-flight async copies landed in LDS
    __syncthreads();       // every wave's copies for chunk c are now visible

    if (c + 1 < N_CHUNKS)  // prefetch next chunk into the other buffer
      fill_chunk(lds + ((c + 1) & 1) * CHUNK_ELEMS, x, (c + 1) * CI_CHUNK,
                 b, y, x0, tid);

    const float* __restrict__ bbuf = lds + (c & 1) * CHUNK_ELEMS;
    const int kc0 = c * CHUNK_K;
    const int ciBase = c * CI_CHUNK;

    for (int k0 = kc0; k0 < kc0 + CHUNK_K; k0 += 4) {
      const int k1 = k0 + half * 2;
      const int k2 = k1 + 1;

      // A fragment: weights, M = co (lane 0-15), K pair per half
      v2f a;
      a.x = W_in[wrow + k1];
      a.y = W_in[wrow + k2];

      // B fragment addresses in LDS: decompose k -> (ci, ky, kx)
      const int ci1 = k1 / 9, r1 = k1 - ci1 * 9;
      const int ky1 = r1 / 3, kx1 = r1 - ky1 * 3;
      const int ci2 = k2 / 9, r2 = k2 - ci2 * 9;
      const int ky2 = r2 / 3, kx2 = r2 - ky2 * 3;
      const int o1 = (ci1 - ciBase) * 198 + ky1 * 66 + kx1 + l16;
      const int o2 = (ci2 - ciBase) * 198 + ky2 * 66 + kx2 + l16;

      v2f bt;
      bt.x = bbuf[o1 +  0]; bt.y = bbuf[o2 +  0];
      acc0 = WMMA_F32X4(a, bt, acc0);
      bt.x = bbuf[o1 + 16]; bt.y = bbuf[o2 + 16];
      acc1 = WMMA_F32X4(a, bt, acc1);
      bt.x = bbuf[o1 + 32]; bt.y = bbuf[o2 + 32];
      acc2 = WMMA_F32X4(a, bt, acc2);
      bt.x = bbuf[o1 + 48]; bt.y = bbuf[o2 + 48];
      acc3 = WMMA_F32X4(a, bt, acc3);
    }
    // No bottom barrier needed: fill for chunk c+2 (buffer c&1) is only
    // issued after the next top-of-loop barrier, which all threads reach
    // only after finishing the reads above.
  }

  // C/D layout: VGPR j -> M = j + 8*half, N = lane&15. N = pixel -> coalesced.
  #pragma unroll
  for (int j = 0; j < 8; ++j) {
    const int co = coBase + j + 8 * half;
    const size_t base = (((size_t)(b * 128 + co)) << 14) + ((size_t)y << 7) + x0 + l16;
    A[base +  0] = acc0[j];
    A[base + 16] = acc1[j];
    A[base + 32] = acc2[j];
    A[base + 48] = acc3[j];
  }
}

// ---------------------------------------------------------------------------
// Kernel 2: depthwise 4x4 conv, pad 3 (output 131x131), + bias + SiLU.
// One thread per output element: 8*128*131*131 = 17,572,864.
// ---------------------------------------------------------------------------
__launch_bounds__(256)
__global__ void dwconv_silu(const float* __restrict__ A,
                            const float* __restrict__ Wc,
                            const float* __restrict__ bc,
                            float* __restrict__ xh)
{
  const long idx = (long)blockIdx.x * 256 + threadIdx.x;
  if (idx >= (long)8 * 128 * 131 * 131) return;

  int t  = (int)(idx / 131);
  const int ox = (int)(idx - (long)t * 131);
  const int oy = t % 131; t /= 131;
  const int c  = t & 127;
  const int b  = t >> 7;

  const float* __restrict__ w  = Wc + c * 16;                    // W_conv[c,0,:,:]
  const float* __restrict__ ap = A + (((size_t)(b * 128 + c)) << 14);

  float s = bc[c];
  #pragma unroll
  for (int ky = 0; ky < 4; ++ky) {
    const int iy = oy + ky - 3;
    if (iy < 0 || iy >= 128) continue;
    #pragma unroll
    for (int kx = 0; kx < 4; ++kx) {
      const int ix = ox + kx - 3;
      if (ix < 0 || ix >= 128) continue;
      s += ap[(iy << 7) + ix] * w[ky * 4 + kx];
    }
  }
  const float sig = 1.0f / (1.0f + __expf(-s));
  xh[idx] = s * sig;
}

// ---------------------------------------------------------------------------
// Kernel 3: out[r,e] = sum_d xh_flat[r*32+d] * W_out[e*32+d]   (549,152 rows)
// One wave per 16-row tile; 2 accumulators cover e = 0..31. K=32 -> 8 WMMAs.
// ---------------------------------------------------------------------------
__launch_bounds__(256)
__global__ void headgemm_wmma(const float* __restrict__ xh,
                              const float* __restrict__ Wout,
                              float* __restrict__ out,
                              long nrows)
{
  const int lane = threadIdx.x & 31;
  const int wave = threadIdx.x >> 5;
  const long R0  = ((long)blockIdx.x * 8 + wave) * 16;
  if (R0 >= nrows) return;

  const int half = lane >> 4;
  const int l16  = lane & 15;

  const float* __restrict__ arow = xh + (R0 + l16) * 32;   // A: M = row
  v8f acc0 = {}, acc1 = {};

  #pragma unroll
  for (int k0 = 0; k0 < 32; k0 += 4) {
    const int kb = k0 + half * 2;
    v2f a;
    a.x = arow[kb];
    a.y = arow[kb + 1];
    v2f b0, b1;                                   // B[k][e] = W_out[e*32+k]
    b0.x = Wout[l16 * 32 + kb];
    b0.y = Wout[l16 * 32 + kb + 1];
    b1.x = Wout[(16 + l16) * 32 + kb];
    b1.y = Wout[(16 + l16) * 32 + kb + 1];
    acc0 = WMMA_F32X4(a, b0, acc0);
    acc1 = WMMA_F32X4(a, b1, acc1);
  }

  #pragma unroll
  for (int j = 0; j < 8; ++j) {
    const long row = R0 + j + 8 * half;
    out[row * 32 + l16]      = acc0[j];
    out[row * 32 + 16 + l16] = acc1[j];
  }
}

// ---------------------------------------------------------------------------
extern "C" void kernel_launch(void* const* d_in, const int* in_sizes, int n_in,
                              void* d_out, int out_size, void* d_ws, size_t ws_size,
                              hipStream_t stream) {
  const float* x      = (const float*)d_in[0];
  const float* W_in   = (const float*)d_in[1];
  const float* W_conv = (const float*)d_in[2];
  const float* b_conv = (const float*)d_in[3];
  // d_in[4] dt_bias, d_in[5] A_log, d_in[6] D: dead in the reference output.
  const float* W_out  = (const float*)d_in[7];

  float* A   = (float*)d_ws;                                  // 67,108,864 B
  float* xh  = (float*)((char*)d_ws + (size_t)16777216 * 4);  // 70,291,456 B
  float* out = (float*)d_out;

  // conv1: 131072 pixels / 64 per block row-segment; co split in 2 tiles of 64
  dim3 g1(2048, 2);
  conv1_wmma<<<g1, 128, 0, stream>>>(x, W_in, A);

  // depthwise 4x4 + bias + SiLU: 17,572,864 elements / 256
  dwconv_silu<<<68644, 256, 0, stream>>>(A, W_conv, b_conv, xh);

  // head GEMM: 549,152 rows, 128 rows per block (8 waves x 16)
  const long nrows = 549152;
  headgemm_wmma<<<4291, 256, 0, stream>>>(xh, W_out, out, nrows);
}